// DigitCaps_83494164234759
// MI455X (gfx1250) — compile-verified
//
#include <hip/hip_runtime.h>
#include <math.h>

#define NN 2592   // 32*9*9 routes
#define LL 64     // output capsule dim
#define CC 16     // input capsule dim
#define BB 128    // batch

typedef __attribute__((ext_vector_type(2))) float v2f;
typedef __attribute__((ext_vector_type(8))) float v8f;

// ---------------------------------------------------------------------------
// Kernel 1: u_hat[b,n,l] = sum_c W[n,l,c] * x[b,n,c]  via V_WMMA_F32_16X16X4_F32
// One wave = one route n, one 16-row batch block, all 64 columns (4 D-tiles).
// A (16x4 f32): lanes 0-15 hold M=0..15; VGPR0 = K+0 / K+2 (lane half),
// VGPR1 = K+1 / K+3  -> per-lane float2 at c = kk + 2*hi.
// B (4x16 f32): mirrored layout -> per-lane float2 from W[n, l=lo, c=kk+2*hi].
// ---------------------------------------------------------------------------
__global__ __launch_bounds__(256) void k_uhat_wmma(const float* __restrict__ x,
                                                   const float* __restrict__ W,
                                                   float* __restrict__ u_hat) {
  const int n    = blockIdx.x;
  const int wave = threadIdx.x >> 5;
  const int lane = threadIdx.x & 31;
  const int lo   = lane & 15;
  const int hi   = lane >> 4;
  const int b0   = wave << 4;                 // 8 waves * 16 rows = 128 = B

  const float* xrow = x + ((size_t)(b0 + lo) * NN + n) * CC;  // x[b0+lo, n, :]
  const float* wn   = W + (size_t)n * LL * CC;                // W[n, :, :]

  v8f acc0 = {}, acc1 = {}, acc2 = {}, acc3 = {};
  #pragma unroll
  for (int kk = 0; kk < CC; kk += 4) {
    v2f a   = *(const v2f*)(xrow + kk + 2 * hi);
    v2f bq0 = *(const v2f*)(wn + (size_t)(0 * 16 + lo) * CC + kk + 2 * hi);
    v2f bq1 = *(const v2f*)(wn + (size_t)(1 * 16 + lo) * CC + kk + 2 * hi);
    v2f bq2 = *(const v2f*)(wn + (size_t)(2 * 16 + lo) * CC + kk + 2 * hi);
    v2f bq3 = *(const v2f*)(wn + (size_t)(3 * 16 + lo) * CC + kk + 2 * hi);
    acc0 = __builtin_amdgcn_wmma_f32_16x16x4_f32(false, a, false, bq0, (short)0, acc0, false, false);
    acc1 = __builtin_amdgcn_wmma_f32_16x16x4_f32(false, a, false, bq1, (short)0, acc1, false, false);
    acc2 = __builtin_amdgcn_wmma_f32_16x16x4_f32(false, a, false, bq2, (short)0, acc2, false, false);
    acc3 = __builtin_amdgcn_wmma_f32_16x16x4_f32(false, a, false, bq3, (short)0, acc3, false, false);
  }

  // D layout: VGPR r, lanes 0-15 -> (M=r, N=lane), lanes 16-31 -> (M=r+8, N=lane-16)
  #pragma unroll
  for (int r = 0; r < 8; ++r) {
    float* dst = u_hat + ((size_t)(b0 + r + 8 * hi) * NN + n) * LL + lo;
    dst[0]  = acc0[r];
    dst[16] = acc1[r];
    dst[32] = acc2[r];
    dst[48] = acc3[r];
  }
}

// ---------------------------------------------------------------------------
// Zero routing logits (ws is NOT re-poisoned between replays -> must re-init)
// ---------------------------------------------------------------------------
__global__ void k_zero(float* __restrict__ p, int nelem) {
  int i = blockIdx.x * blockDim.x + threadIdx.x;
  if (i < nelem) p[i] = 0.0f;
}

// ---------------------------------------------------------------------------
// Softmax over N=2592 logits (single block; tiny)
// ---------------------------------------------------------------------------
__global__ __launch_bounds__(256) void k_softmax(const float* __restrict__ b_ij,
                                                 float* __restrict__ c_ij) {
  __shared__ float red[256];
  const int t = threadIdx.x;
  float m = -3.402823466e38f;
  for (int n = t; n < NN; n += 256) m = fmaxf(m, b_ij[n]);
  red[t] = m; __syncthreads();
  for (int s2 = 128; s2 > 0; s2 >>= 1) {
    if (t < s2) red[t] = fmaxf(red[t], red[t + s2]);
    __syncthreads();
  }
  const float mx = red[0];
  __syncthreads();
  float sum = 0.0f;
  for (int n = t; n < NN; n += 256) sum += __expf(b_ij[n] - mx);
  red[t] = sum; __syncthreads();
  for (int s2 = 128; s2 > 0; s2 >>= 1) {
    if (t < s2) red[t] += red[t + s2];
    __syncthreads();
  }
  const float inv = 1.0f / red[0];
  for (int n = t; n < NN; n += 256) c_ij[n] = __expf(b_ij[n] - mx) * inv;
}

// ---------------------------------------------------------------------------
// s_j[b,:] = sum_n c[n]*u_hat[b,n,:] ;  v = squash(s). L2-resident re-read.
// ---------------------------------------------------------------------------
__global__ __launch_bounds__(256) void k_route_sv(const float* __restrict__ u_hat,
                                                  const float* __restrict__ c_ij,
                                                  float* __restrict__ v,
                                                  float* __restrict__ v_out,
                                                  int write_out) {
  __shared__ float red[256];
  const int b = blockIdx.x;
  const int t = threadIdx.x;
  const int l = t & 63;
  const int g = t >> 6;
  const float* uh = u_hat + (size_t)b * NN * LL;
  float acc = 0.0f;
  for (int n = g; n < NN; n += 4) acc += c_ij[n] * uh[(size_t)n * LL + l];
  red[t] = acc; __syncthreads();
  if (t < 64) {
    float s  = red[l] + red[l + 64] + red[l + 128] + red[l + 192];
    float sq = s * s;
    float vv = sq * s / ((1.0f + sq) * sqrtf(sq));   // faithful squash
    v[b * LL + l] = vv;
    if (write_out) v_out[b * LL + l] = vv;
  }
}

// ---------------------------------------------------------------------------
// b_ij[n] += mean_b( dot(u_hat[b,n,:], v[b,:]) )
// ---------------------------------------------------------------------------
__global__ __launch_bounds__(256) void k_route_a(const float* __restrict__ u_hat,
                                                 const float* __restrict__ v,
                                                 float* __restrict__ b_ij) {
  __shared__ float red[256];
  const int n = blockIdx.x;
  const int t = threadIdx.x;
  float acc = 0.0f;
  for (int i = t; i < BB * LL; i += 256) {
    int b = i >> 6, l = i & 63;
    acc += u_hat[((size_t)b * NN + n) * LL + l] * v[i];
  }
  red[t] = acc; __syncthreads();
  for (int s2 = 128; s2 > 0; s2 >>= 1) {
    if (t < s2) red[t] += red[t + s2];
    __syncthreads();
  }
  if (t == 0) b_ij[n] += red[0] * (1.0f / BB);
}

// ---------------------------------------------------------------------------
// u_norm[b,n] = c[n] * ||u_hat[b,n,:]||   (c>=0, so ||c*u|| = c*||u||)
// One wave per row; float2 per lane + shuffle-xor reduction (wave32).
// ---------------------------------------------------------------------------
__global__ __launch_bounds__(256) void k_unorm(const float* __restrict__ u_hat,
                                               const float* __restrict__ c_ij,
                                               float* __restrict__ u_norm) {
  const int wave = threadIdx.x >> 5;
  const int lane = threadIdx.x & 31;
  const int row  = blockIdx.x * 8 + wave;          // row = b*NN + n, exact tiling
  const float* p = u_hat + (size_t)row * LL + lane * 2;
  float a0 = p[0], a1 = p[1];
  float s = a0 * a0 + a1 * a1;
  #pragma unroll
  for (int m = 16; m > 0; m >>= 1) s += __shfl_xor(s, m, 32);
  if (lane == 0) {
    int n = row % NN;
    u_norm[row] = c_ij[n] * sqrtf(s);
  }
}

// ---------------------------------------------------------------------------
// ConvTranspose2d(32->64, k=2, s=2, p=1): each output pixel has exactly one
// kernel tap: kh=(h+1)&1, ih=(h+1)>>1 (same for w). 32-wide dot per pixel.
// out[b,oc,h,w] = bias[oc] + sum_ic u_norm[b,ic,ih,iw]*w[ic,oc,kh,kw]
// ---------------------------------------------------------------------------
__global__ __launch_bounds__(256) void k_convt(const float* __restrict__ u_norm,
                                               const float* __restrict__ cw,
                                               const float* __restrict__ cb,
                                               float* __restrict__ out) {
  const int t  = blockIdx.x * 256 + threadIdx.x;   // 128*64*16*16 = 2,097,152
  const int w  = t & 15;
  const int h  = (t >> 4) & 15;
  const int oc = (t >> 8) & 63;
  const int b  = t >> 14;
  const int kh = (h + 1) & 1, ih = (h + 1) >> 1;
  const int kw = (w + 1) & 1, iw = (w + 1) >> 1;
  const float* un = u_norm + (size_t)b * NN + ih * 9 + iw;  // stride 81 over ic
  const float* wp = cw + oc * 4 + kh * 2 + kw;              // stride 256 over ic
  float acc = cb[oc];
  #pragma unroll 8
  for (int ic = 0; ic < 32; ++ic)
    acc = fmaf(un[ic * 81], wp[ic * 256], acc);
  out[t] = acc;
}

// ---------------------------------------------------------------------------
extern "C" void kernel_launch(void* const* d_in, const int* in_sizes, int n_in,
                              void* d_out, int out_size, void* d_ws, size_t ws_size,
                              hipStream_t stream) {
  const float* x  = (const float*)d_in[0];   // (B, N, C)
  const float* W  = (const float*)d_in[1];   // (1, N, 1, L, C)
  const float* cw = (const float*)d_in[2];   // (32, 64, 2, 2)
  const float* cb = (const float*)d_in[3];   // (64,)
  float* out = (float*)d_out;                // v_j (B*64) ++ out_img (B*64*16*16)

  float* ws     = (float*)d_ws;
  float* u_hat  = ws;                                    // B*NN*LL = 21,233,664 f
  float* b_ij   = u_hat + (size_t)BB * NN * LL;          // NN
  float* c_ij   = b_ij + NN;                             // NN
  float* v      = c_ij + NN;                             // B*LL
  float* u_nrm  = v + BB * LL;                           // B*NN

  (void)in_sizes; (void)n_in; (void)out_size; (void)ws_size;

  k_zero<<<(NN + 255) / 256, 256, 0, stream>>>(b_ij, NN);
  k_uhat_wmma<<<NN, 256, 0, stream>>>(x, W, u_hat);

  for (int it = 0; it < 3; ++it) {
    k_softmax<<<1, 256, 0, stream>>>(b_ij, c_ij);
    k_route_sv<<<BB, 256, 0, stream>>>(u_hat, c_ij, v, out, (it == 2) ? 1 : 0);
    if (it < 2) k_route_a<<<NN, 256, 0, stream>>>(u_hat, v, b_ij);
  }

  k_unorm<<<(BB * NN) / 8, 256, 0, stream>>>(u_hat, c_ij, u_nrm);
  k_convt<<<(BB * 64 * 16 * 16) / 256, 256, 0, stream>>>(u_nrm, cw, cb, out + BB * LL);
}